// QueryAndGroup_12945031430503
// MI455X (gfx1250) — compile-verified
//
#include <hip/hip_runtime.h>

// Problem constants (match reference setup_inputs)
#define BB   4
#define NP   8192
#define NQ   2048
#define KK   32
#define CC   64
#define CT   64
#define NCHUNK (NP / 32)          // 256 chunks of 32 points (2 wmma tiles each)
#define RAD2 0.04f                // 0.2^2

typedef __attribute__((ext_vector_type(2))) float v2f;
typedef __attribute__((ext_vector_type(8))) float v8f;

// ---------------------------------------------------------------------------
// Phase 0: pack points as (x, y, z, |p|^2) for single-b128 loads in the wmma loop
// ---------------------------------------------------------------------------
__global__ void pack_points_kernel(const float* __restrict__ coords,
                                   float4* __restrict__ pts4, int total) {
    int i = blockIdx.x * blockDim.x + threadIdx.x;
    if (i < total) {
        float x = coords[3 * i + 0];
        float y = coords[3 * i + 1];
        float z = coords[3 * i + 2];
        float4 p; p.x = x; p.y = y; p.z = z; p.w = x * x + y * y + z * z;
        pts4[i] = p;
    }
}

// 8-to-1 mux of wave-uniform ballot results (7 v_cndmask)
__device__ __forceinline__ unsigned mux8(unsigned sel,
        unsigned m0, unsigned m1, unsigned m2, unsigned m3,
        unsigned m4, unsigned m5, unsigned m6, unsigned m7) {
    unsigned s0 = (sel & 1u) ? m1 : m0;
    unsigned s1 = (sel & 1u) ? m3 : m2;
    unsigned s2 = (sel & 1u) ? m5 : m4;
    unsigned s3 = (sel & 1u) ? m7 : m6;
    unsigned t0 = (sel & 2u) ? s1 : s0;
    unsigned t1 = (sel & 2u) ? s3 : s2;
    return (sel & 4u) ? t1 : t0;
}

// ---------------------------------------------------------------------------
// Phase 1: ball query via V_WMMA_F32_16X16X4_F32
//   A row m  = (qx, qy, qz, 1)           -> 16 queries per wave
//   B col n  = (-2px, -2py, -2pz, |p|^2) -> 16 points per tile
//   D[m][n]  = |p|^2 - 2 q.p  ;  in-ball iff D < r^2 - |q|^2
// One wave handles 16 queries; streams 32-point chunks (2 wmma tiles) in index
// order, ballots the in-ball masks, fuses them into a 32-bit word per query,
// and lanes 0..15 extract the first-K indices in order with early exit.
// ---------------------------------------------------------------------------
__global__ __launch_bounds__(256) void ball_query_wmma_kernel(
        const float4* __restrict__ pts4,
        const float*  __restrict__ queries,
        int*          __restrict__ idxs) {
    const int lane = threadIdx.x & 31;
    const int wave = blockIdx.x * 8 + (threadIdx.x >> 5);   // 512 waves total
    const int b    = wave / (NQ / 16);
    const int q0   = (wave % (NQ / 16)) * 16;
    const int m    = lane & 15;

    // Per-lane query (lanes 16..31 mirror lanes 0..15's queries)
    const float* qp = queries + ((size_t)b * NQ + q0 + m) * 3;
    float qx = qp[0], qy = qp[1], qz = qp[2];
    float myth = RAD2 - (qx * qx + qy * qy + qz * qz);      // row threshold

    // A fragment: VGPR0 = K0 (lanes 0-15) / K2 (16-31); VGPR1 = K1 / K3
    v2f a;
    if (lane < 16) { a.x = qx; a.y = qy; }
    else           { a.x = qz; a.y = 1.0f; }

    // Row thresholds for the 8 D VGPRs: lanes<16 compare rows 0..7,
    // lanes>=16 compare rows 8..15.  Distribute via cross-lane shuffle.
    const int thOff = (lane >= 16) ? 8 : 0;
    float th0 = __shfl(myth, 0 + thOff, 32);
    float th1 = __shfl(myth, 1 + thOff, 32);
    float th2 = __shfl(myth, 2 + thOff, 32);
    float th3 = __shfl(myth, 3 + thOff, 32);
    float th4 = __shfl(myth, 4 + thOff, 32);
    float th5 = __shfl(myth, 5 + thOff, 32);
    float th6 = __shfl(myth, 6 + thOff, 32);
    float th7 = __shfl(myth, 7 + thOff, 32);

    const float4* pb = pts4 + (size_t)b * NP;
    int* out = idxs + ((size_t)b * NQ + q0 + m) * KK;       // lanes 0..15 own a query
    int cnt = 0;
    int firstIdx = 0;
    const unsigned sel = (unsigned)(m & 7);

    for (int j = 0; j < NCHUNK; ++j) {
        // Two 16-point tiles per chunk
        float4 p0 = pb[j * 32 + m];                          // global_load_b128
        float4 p1 = pb[j * 32 + 16 + m];
        if (j + 1 < NCHUNK)
            __builtin_prefetch(&pb[(j + 1) * 32 + m], 0, 1); // global_prefetch_b8

        v2f bv0, bv1;
        if (lane < 16) {
            bv0.x = -2.0f * p0.x; bv0.y = -2.0f * p0.y;
            bv1.x = -2.0f * p1.x; bv1.y = -2.0f * p1.y;
        } else {
            bv0.x = -2.0f * p0.z; bv0.y = p0.w;
            bv1.x = -2.0f * p1.z; bv1.y = p1.w;
        }

        v8f c0 = {}, c1 = {};
        c0 = __builtin_amdgcn_wmma_f32_16x16x4_f32(
                false, a, false, bv0, (short)0, c0, false, false);
        c1 = __builtin_amdgcn_wmma_f32_16x16x4_f32(
                false, a, false, bv1, (short)0, c1, false, false);

        // D VGPR r: lanes 0-15 = (row r, col=lane); lanes 16-31 = (row r+8, col=lane-16)
        // One ballot per VGPR -> bits[0:15] = row r mask, bits[16:31] = row r+8 mask.
        unsigned a0 = __builtin_amdgcn_ballot_w32(c0[0] < th0);
        unsigned a1 = __builtin_amdgcn_ballot_w32(c0[1] < th1);
        unsigned a2 = __builtin_amdgcn_ballot_w32(c0[2] < th2);
        unsigned a3 = __builtin_amdgcn_ballot_w32(c0[3] < th3);
        unsigned a4 = __builtin_amdgcn_ballot_w32(c0[4] < th4);
        unsigned a5 = __builtin_amdgcn_ballot_w32(c0[5] < th5);
        unsigned a6 = __builtin_amdgcn_ballot_w32(c0[6] < th6);
        unsigned a7 = __builtin_amdgcn_ballot_w32(c0[7] < th7);
        unsigned b0 = __builtin_amdgcn_ballot_w32(c1[0] < th0);
        unsigned b1 = __builtin_amdgcn_ballot_w32(c1[1] < th1);
        unsigned b2 = __builtin_amdgcn_ballot_w32(c1[2] < th2);
        unsigned b3 = __builtin_amdgcn_ballot_w32(c1[3] < th3);
        unsigned b4 = __builtin_amdgcn_ballot_w32(c1[4] < th4);
        unsigned b5 = __builtin_amdgcn_ballot_w32(c1[5] < th5);
        unsigned b6 = __builtin_amdgcn_ballot_w32(c1[6] < th6);
        unsigned b7 = __builtin_amdgcn_ballot_w32(c1[7] < th7);

        // Ballot results are wave-uniform: mux the one this lane's query needs,
        // then fuse both tiles' 16-bit row masks into one 32-bit word.
        unsigned wa = mux8(sel, a0, a1, a2, a3, a4, a5, a6, a7);
        unsigned wb = mux8(sel, b0, b1, b2, b3, b4, b5, b6, b7);
        unsigned lo = (m < 8) ? (wa & 0xFFFFu) : (wa >> 16);
        unsigned hi = (m < 8) ? (wb & 0xFFFFu) : (wb >> 16);
        unsigned bits = lo | (hi << 16);

        if (lane < 16) {
            int base = j * 32;
            while (bits && cnt < KK) {
                int t = __builtin_ctz(bits);
                bits &= bits - 1u;
                int pi = base + t;
                if (cnt == 0) firstIdx = pi;
                out[cnt++] = pi;
            }
        }
        // Uniform early exit once all 16 queries in this wave are full.
        if (__builtin_amdgcn_ballot_w32(lane < 16 && cnt < KK) == 0u)
            break;
    }

    if (lane < 16) {
        // Pad with first valid index (or 0 if none) — matches reference semantics.
        for (int kk = cnt; kk < KK; ++kk) out[kk] = firstIdx;
    }
}

// ---------------------------------------------------------------------------
// Phase 2: gather + group.  One block per (b, q).  Output-coalesced stores;
// gather reads hit L2 (features/t_embed are 2 MB per batch, L2 is 192 MB).
// out1 = [B, 3+C, Nq, K]   out2 = [B, Ct, Nq, K]
// ---------------------------------------------------------------------------
__global__ __launch_bounds__(256) void gather_group_kernel(
        const float* __restrict__ coords,
        const float* __restrict__ feats,
        const float* __restrict__ tembed,
        const float* __restrict__ queries,
        const int*   __restrict__ idxs,
        float* __restrict__ out1,
        float* __restrict__ out2) {
    const int bq = blockIdx.x;             // 0 .. B*NQ-1
    const int b  = bq / NQ;
    const int q  = bq % NQ;

    __shared__ int   s_idx[KK];
    __shared__ float s_q[3];

    const int t = threadIdx.x;
    if (t < KK) s_idx[t] = idxs[(size_t)bq * KK + t];
    if (t < 3)  s_q[t]   = queries[(size_t)bq * 3 + t];
    __syncthreads();

    const int k  = t & 31;
    const int c0 = t >> 5;                 // 8 channel groups

    const size_t qkoff = (size_t)q * KK + k;
    float* o1 = out1 + (size_t)b * (3 + CC) * NQ * KK + qkoff;
    float* o2 = out2 + (size_t)b * CT * NQ * KK + qkoff;

    // Centered coords -> channels 0..2 of out1
    if (c0 < 3) {
        float v = coords[((size_t)b * NP + s_idx[k]) * 3 + c0] - s_q[c0];
        o1[(size_t)c0 * NQ * KK] = v;
    }
    // Feature gather -> channels 3..66 of out1
    const float* fb = feats + (size_t)b * CC * NP;
    #pragma unroll
    for (int c = 0; c < CC / 8; ++c) {
        int ch = c0 + c * 8;
        o1[(size_t)(3 + ch) * NQ * KK] = fb[(size_t)ch * NP + s_idx[k]];
    }
    // t_embed gather -> out2
    const float* tb = tembed + (size_t)b * CT * NP;
    #pragma unroll
    for (int c = 0; c < CT / 8; ++c) {
        int ch = c0 + c * 8;
        o2[(size_t)ch * NQ * KK] = tb[(size_t)ch * NP + s_idx[k]];
    }
}

// ---------------------------------------------------------------------------
extern "C" void kernel_launch(void* const* d_in, const int* in_sizes, int n_in,
                              void* d_out, int out_size, void* d_ws, size_t ws_size,
                              hipStream_t stream) {
    const float* coords  = (const float*)d_in[0];  // [B,Np,3]
    const float* feats   = (const float*)d_in[1];  // [B,C,Np]
    const float* tembed  = (const float*)d_in[2];  // [B,Ct,Np]
    const float* queries = (const float*)d_in[3];  // [B,Nq,3]

    float* out1 = (float*)d_out;                                   // [B,67,Nq,K]
    float* out2 = out1 + (size_t)BB * (3 + CC) * NQ * KK;          // [B,64,Nq,K]

    // Workspace layout: packed points, then idxs
    float4* pts4 = (float4*)d_ws;                                  // 512 KB
    int*    idxs = (int*)((char*)d_ws + sizeof(float4) * (size_t)BB * NP); // 1 MB

    const int totalPts = BB * NP;
    pack_points_kernel<<<(totalPts + 255) / 256, 256, 0, stream>>>(coords, pts4, totalPts);

    // 512 waves of 16 queries each; 8 waves per block
    ball_query_wmma_kernel<<<(BB * NQ / 16) / 8, 256, 0, stream>>>(pts4, queries, idxs);

    gather_group_kernel<<<BB * NQ, 256, 0, stream>>>(coords, feats, tembed,
                                                     queries, idxs, out1, out2);
}